// SimplifiedMambaBlock_9698036155145
// MI455X (gfx1250) — compile-verified
//
#include <hip/hip_runtime.h>
#include <hip/hip_bf16.h>
#include <cmath>

// ---------------------------------------------------------------------------
// SimplifiedMambaBlock for MI455X (gfx1250, wave32, WMMA).
// GEMMs: V_WMMA_F32_16X16X4_F32 (exact fp32 semantics = fp32 JAX reference),
// 2x2 register blocking per wave (32x32 tile), 64x128 block tile, LDS-staged
// transposed operands with padding for conflict-free fragment reads.
// Scan: lane-per-state (16 lanes / (b,d) channel), 2048 concurrent waves.
// ---------------------------------------------------------------------------

typedef __attribute__((ext_vector_type(2))) float v2f;
typedef __attribute__((ext_vector_type(8))) float v8f;

#define BM 64
#define BN 128
#define KC 32
#define LDA (BM + 4)
#define LDB (BN + 4)

__device__ __forceinline__ float silu_f(float v) {
    return v / (1.0f + __expf(-v));
}

__device__ __forceinline__ v8f wmma_f32(v2f a, v2f b, v8f c) {
    return __builtin_amdgcn_wmma_f32_16x16x4_f32(
        false, a, false, b, (short)0, c, false, false);
}

// D[M,N] = A[M,K] @ W[N,K]^T + bias[N]  (+ epilogue per `mode`)
// mode 0: bias only | mode 1: bias+softplus | mode 2: bias+residual[M,N]
__global__ __launch_bounds__(256)
void gemm_wmma_f32_kernel(const float* __restrict__ A, const float* __restrict__ W,
                          const float* __restrict__ bias, const float* __restrict__ resid,
                          float* __restrict__ C, int M, int N, int K, int mode)
{
    __shared__ float AsT[KC][LDA];   // A tile, transposed: AsT[k][m]
    __shared__ float Bs[KC][LDB];    // W tile, transposed: Bs[k][n]

    const int tid  = threadIdx.x;
    const int wave = tid >> 5;
    const int lane = tid & 31;
    const int half = lane >> 4;      // 0: lanes 0-15 (K=0/1), 1: lanes 16-31 (K=2/3)
    const int l16  = lane & 15;
    const int wr   = wave >> 2;      // 0..1 : 32-row wave tile within 64-row block
    const int wc   = wave & 3;       // 0..3 : 32-col wave tile within 128-col block
    const int m0   = blockIdx.y * BM;
    const int n0   = blockIdx.x * BN;

    v8f acc00 = {}, acc01 = {}, acc10 = {}, acc11 = {};

    for (int k0 = 0; k0 < K; k0 += KC) {
        // --- stage A (BM x KC) transposed: 512 float4, 2 per thread ---
        #pragma unroll
        for (int rep = 0; rep < 2; ++rep) {
            const int idx = tid + rep * 256;
            const int mm  = idx >> 3;          // 0..63
            const int kk  = (idx & 7) << 2;    // 0,4,...,28
            const float4 av = *(const float4*)(A + (size_t)(m0 + mm) * K + (k0 + kk));
            AsT[kk + 0][mm] = av.x;
            AsT[kk + 1][mm] = av.y;
            AsT[kk + 2][mm] = av.z;
            AsT[kk + 3][mm] = av.w;
        }
        // --- stage W (BN x KC) transposed: 1024 float4, 4 per thread ---
        #pragma unroll
        for (int rep = 0; rep < 4; ++rep) {
            const int idx = tid + rep * 256;
            const int nn  = idx >> 3;          // 0..127
            const int kk  = (idx & 7) << 2;    // 0,4,...,28
            float4 wv = {0.f, 0.f, 0.f, 0.f};
            if (n0 + nn < N)
                wv = *(const float4*)(W + (size_t)(n0 + nn) * K + (k0 + kk));
            Bs[kk + 0][nn] = wv.x;
            Bs[kk + 1][nn] = wv.y;
            Bs[kk + 2][nn] = wv.z;
            Bs[kk + 3][nn] = wv.w;
        }
        __syncthreads();

        // --- 8 K-steps x 4 WMMA (2x2 register block) = 32 WMMA per chunk ---
        #pragma unroll
        for (int ks = 0; ks < KC; ks += 4) {
            const int kA = ks + half * 2;
            v2f a0, a1, b0, b1;
            a0[0] = AsT[kA + 0][wr * 32 + l16];
            a0[1] = AsT[kA + 1][wr * 32 + l16];
            a1[0] = AsT[kA + 0][wr * 32 + 16 + l16];
            a1[1] = AsT[kA + 1][wr * 32 + 16 + l16];
            b0[0] = Bs[kA + 0][wc * 32 + l16];
            b0[1] = Bs[kA + 1][wc * 32 + l16];
            b1[0] = Bs[kA + 0][wc * 32 + 16 + l16];
            b1[1] = Bs[kA + 1][wc * 32 + 16 + l16];
            acc00 = wmma_f32(a0, b0, acc00);
            acc01 = wmma_f32(a0, b1, acc01);
            acc10 = wmma_f32(a1, b0, acc10);
            acc11 = wmma_f32(a1, b1, acc11);
        }
        __syncthreads();
    }

    // --- epilogue: C/D layout -> VGPR v holds (M = v + 8*half, N = l16) ---
    #pragma unroll
    for (int mi = 0; mi < 2; ++mi) {
        const v8f* accRow = (mi == 0) ? &acc00 : &acc10;  // [ni] selected below
        #pragma unroll
        for (int ni = 0; ni < 2; ++ni) {
            const v8f accv = (mi == 0) ? (ni == 0 ? acc00 : acc01)
                                       : (ni == 0 ? acc10 : acc11);
            const int nOut = n0 + wc * 32 + ni * 16 + l16;
            if (nOut < N) {
                const float bval = bias[nOut];
                const int mBase = m0 + wr * 32 + mi * 16 + half * 8;
                #pragma unroll
                for (int v = 0; v < 8; ++v) {
                    const int mOut = mBase + v;
                    float r = accv[v] + bval;
                    if (mode == 1) {
                        r = (r > 20.f) ? r : log1pf(__expf(r));   // softplus
                    } else if (mode == 2) {
                        r += resid[(size_t)mOut * N + nOut];      // residual
                    }
                    C[(size_t)mOut * (size_t)N + nOut] = r;
                }
            }
        }
        (void)accRow;
    }
}

// causal depthwise conv1d (k=4, left pad 3) + SiLU.
// reads x_ssm = first D columns of xproj (stride 2D), writes xconv (B,L,D)
__global__ __launch_bounds__(256)
void conv_silu_kernel(const float* __restrict__ xproj, const float* __restrict__ cw,
                      const float* __restrict__ cb, float* __restrict__ xconv,
                      int L, int D, int twoD)
{
    const int idx = blockIdx.x * blockDim.x + threadIdx.x;   // over B*L*D
    const int d  = idx % D;
    const int bl = idx / D;
    const int t  = bl % L;
    const int b  = bl / L;

    float s = cb[d];
    #pragma unroll
    for (int j = 0; j < 4; ++j) {
        const int tt = t - 3 + j;
        if (tt >= 0)
            s += cw[d * 4 + j] * xproj[((size_t)b * L + tt) * twoD + d];
    }
    xconv[idx] = silu_f(s);
}

// selective scan. 16 lanes = one (b,d) channel, one state per lane.
// y = (sum_n h_n * C_n + x*D) * silu(gate), gate = xproj[:, D+d]
__global__ __launch_bounds__(256)
void scan_kernel(const float* __restrict__ xconv, const float* __restrict__ dt,
                 const float* __restrict__ bc, const float* __restrict__ xproj,
                 const float* __restrict__ A_log, const float* __restrict__ Dp,
                 float* __restrict__ y, int L, int D, int twoD)
{
    const int grp = threadIdx.x >> 4;                // 16-lane group within block
    const int n   = threadIdx.x & 15;                // state index
    const int ch  = blockIdx.x * 16 + grp;           // channel in [0, B*D)
    const int b   = ch / D;
    const int d   = ch % D;

    const float Aval = -__expf(A_log[d * 16 + n]);
    const float Dd   = Dp[d];
    float h = 0.f;
    const size_t rowB = (size_t)b * L;

    for (int t = 0; t < L; ++t) {
        const size_t r = rowB + t;
        const float xt  = xconv[r * D + d];
        const float dtt = dt[r * D + d];
        const float Bt  = bc[r * 32 + n];
        const float Ct  = bc[r * 32 + 16 + n];

        h = h * __expf(dtt * Aval) + (dtt * Bt) * xt;

        float p = h * Ct;
        p += __shfl_xor(p, 8, 16);
        p += __shfl_xor(p, 4, 16);
        p += __shfl_xor(p, 2, 16);
        p += __shfl_xor(p, 1, 16);

        if (n == 0) {
            const float g = xproj[r * twoD + D + d];
            y[r * D + d] = (p + xt * Dd) * silu_f(g);
        }
    }
}

// row-wise LayerNorm over D=1024 (256 threads x float4 per row)
__global__ __launch_bounds__(256)
void layernorm_kernel(const float* __restrict__ z, const float* __restrict__ g,
                      const float* __restrict__ bta, float* __restrict__ out, int D)
{
    __shared__ float redS[8];
    __shared__ float redQ[8];
    const int row = blockIdx.x;
    const int tid = threadIdx.x;

    const float4 v = *(const float4*)(z + (size_t)row * D + tid * 4);
    float s  = v.x + v.y + v.z + v.w;
    float sq = v.x * v.x + v.y * v.y + v.z * v.z + v.w * v.w;

    #pragma unroll
    for (int off = 16; off > 0; off >>= 1) {
        s  += __shfl_xor(s,  off, 32);
        sq += __shfl_xor(sq, off, 32);
    }
    if ((tid & 31) == 0) { redS[tid >> 5] = s; redQ[tid >> 5] = sq; }
    __syncthreads();

    float ts = 0.f, tq = 0.f;
    #pragma unroll
    for (int i = 0; i < 8; ++i) { ts += redS[i]; tq += redQ[i]; }

    const float mu  = ts / (float)D;
    const float var = tq / (float)D - mu * mu;
    const float inv = rsqrtf(var + 1e-5f);

    float4 o;
    o.x = (v.x - mu) * inv * g[tid * 4 + 0] + bta[tid * 4 + 0];
    o.y = (v.y - mu) * inv * g[tid * 4 + 1] + bta[tid * 4 + 1];
    o.z = (v.z - mu) * inv * g[tid * 4 + 2] + bta[tid * 4 + 2];
    o.w = (v.w - mu) * inv * g[tid * 4 + 3] + bta[tid * 4 + 3];
    *(float4*)(out + (size_t)row * D + tid * 4) = o;
}

extern "C" void kernel_launch(void* const* d_in, const int* in_sizes, int n_in,
                              void* d_out, int out_size, void* d_ws, size_t ws_size,
                              hipStream_t stream) {
    (void)in_sizes; (void)n_in; (void)out_size; (void)ws_size;

    const float* x         = (const float*)d_in[0];   // (4,2048,1024)
    const float* in_proj_w = (const float*)d_in[1];   // (2048,1024)
    const float* in_proj_b = (const float*)d_in[2];   // (2048,)
    const float* conv_w    = (const float*)d_in[3];   // (1024,1,4)
    const float* conv_b    = (const float*)d_in[4];   // (1024,)
    const float* bc_w      = (const float*)d_in[5];   // (32,1024)
    const float* bc_b      = (const float*)d_in[6];   // (32,)
    const float* dt_w      = (const float*)d_in[7];   // (1024,1024)
    const float* dt_b      = (const float*)d_in[8];   // (1024,)
    const float* A_log     = (const float*)d_in[9];   // (1024,16)
    const float* Dvec      = (const float*)d_in[10];  // (1024,)
    const float* out_w     = (const float*)d_in[11];  // (1024,1024)
    const float* out_b     = (const float*)d_in[12];  // (1024,)
    const float* ln_g      = (const float*)d_in[13];  // (1024,)
    const float* ln_b      = (const float*)d_in[14];  // (1024,)

    const int Bb = 4, L = 2048, D = 1024;
    const int M = Bb * L;            // 8192 rows
    const int twoD = 2 * D;

    float* ws    = (float*)d_ws;
    float* xproj = ws;                                   // M * 2D
    float* xconv = xproj + (size_t)M * twoD;             // M * D
    float* dtbuf = xconv + (size_t)M * D;                // M * D
    float* bcbuf = dtbuf + (size_t)M * D;                // M * 32
    float* ybuf  = bcbuf + (size_t)M * 32;               // M * D
    float* zbuf  = ybuf  + (size_t)M * D;                // M * D
    float* outp  = (float*)d_out;

    dim3 blk(256);

    // 1) in_proj: xproj = x @ in_proj_w^T + b            (8192 x 2048)
    gemm_wmma_f32_kernel<<<dim3(twoD / BN, M / BM), blk, 0, stream>>>(
        x, in_proj_w, in_proj_b, nullptr, xproj, M, twoD, D, 0);

    // 2) causal depthwise conv + SiLU -> xconv           (8192 x 1024)
    conv_silu_kernel<<<(M * D) / 256, blk, 0, stream>>>(
        xproj, conv_w, conv_b, xconv, L, D, twoD);

    // 3) dt = softplus(xconv @ dt_w^T + b)               (8192 x 1024)
    gemm_wmma_f32_kernel<<<dim3(D / BN, M / BM), blk, 0, stream>>>(
        xconv, dt_w, dt_b, nullptr, dtbuf, M, D, D, 1);

    // 4) BC = xconv @ bc_w^T + b                         (8192 x 32)
    gemm_wmma_f32_kernel<<<dim3(1, M / BM), blk, 0, stream>>>(
        xconv, bc_w, bc_b, nullptr, bcbuf, M, 32, D, 0);

    // 5) selective scan + gate                            (8192 x 1024)
    scan_kernel<<<(Bb * D) / 16, blk, 0, stream>>>(
        xconv, dtbuf, bcbuf, xproj, A_log, Dvec, ybuf, L, D, twoD);

    // 6) out = y @ out_w^T + b + x (residual)            (8192 x 1024)
    gemm_wmma_f32_kernel<<<dim3(D / BN, M / BM), blk, 0, stream>>>(
        ybuf, out_w, out_b, x, zbuf, M, D, D, 2);

    // 7) LayerNorm -> d_out
    layernorm_kernel<<<M, blk, 0, stream>>>(zbuf, ln_g, ln_b, outp, D);
}